// KGATConv_84756884619934
// MI455X (gfx1250) — compile-verified
//
#include <hip/hip_runtime.h>

#define N_NODES 50000
#define N_EDGES 640000
#define D 128
#define N_TILES (N_NODES / 16)   // 3125, exact
#define LDS_STRIDE 132           // 128 + 4 dword pad -> conflict-free b64 LDS reads

typedef float v2f __attribute__((ext_vector_type(2)));
typedef float v8f __attribute__((ext_vector_type(8)));

// ---------------------------------------------------------------------------
// Kernel 1: zero the h_neighbor accumulator region (first half of d_out).
// ---------------------------------------------------------------------------
__global__ void kgat_zero(float* __restrict__ p, int n4) {
    int i = blockIdx.x * blockDim.x + threadIdx.x;
    if (i < n4) {
        ((float4*)p)[i] = make_float4(0.f, 0.f, 0.f, 0.f);
    }
}

// Hardware no-return float atomic add (avoids any CAS-loop lowering).
__device__ __forceinline__ void gatomic_fadd(float* p, float v) {
    asm volatile("global_atomic_add_f32 %0, %1, off"
                 :
                 : "v"(p), "v"(v)
                 : "memory");
}

// ---------------------------------------------------------------------------
// Kernel 2: edge scatter.  One wave32 per edge; lane l handles 4 features.
//   hn[dst[e]] += nfeat[src[e]] * w[e]
// 82M L2-resident f32 atomics; destination region (25.6 MB) stays in the
// 192 MB L2, so this is L2-atomic-throughput bound — the hardware path.
// ---------------------------------------------------------------------------
__global__ void kgat_scatter(const float* __restrict__ nfeat,
                             const float* __restrict__ w,
                             const int*   __restrict__ src,
                             const int*   __restrict__ dst,
                             float* __restrict__ hn) {
    const int gtid = blockIdx.x * blockDim.x + threadIdx.x;
    const int edge = gtid >> 5;          // wave-uniform
    const int lane = threadIdx.x & 31;
    if (edge >= N_EDGES) return;

    const int   s  = src[edge];
    const int   d  = dst[edge];
    const float we = w[edge];

    const float4 hv = *(const float4*)(nfeat + (size_t)s * D + lane * 4);
    float* outp = hn + (size_t)d * D + lane * 4;

    gatomic_fadd(outp + 0, hv.x * we);
    gatomic_fadd(outp + 1, hv.y * we);
    gatomic_fadd(outp + 2, hv.z * we);
    gatomic_fadd(outp + 3, hv.w * we);
}

// ---------------------------------------------------------------------------
// Kernel 3: bi-interaction GEMM with fp32 WMMA (V_WMMA_F32_16X16X4_F32).
//   out = lrelu((h+hn) @ W1^T) + lrelu((h*hn) @ W2^T)
//
// 256-thread block = 8 waves; each wave owns one 16-row tile.  W1/W2 are
// staged once per block into LDS (2 x 128 x 132 dwords = 132 KB of the WGP's
// 320 KB), then the K-loop feeds WMMA B-fragments with conflict-free
// ds_load_b64 (row stride 132 dwords => bank = (4n + kb + khalf) mod 64,
// each instruction touches all 64 bank slots exactly once).
//
// A layout (16x4 f32, MxK):  lanes 0-15: M=lane,    VGPR0=K[kb+0], VGPR1=K[kb+1]
//                            lanes16-31: M=lane-16, VGPR0=K[kb+2], VGPR1=K[kb+3]
// C/D layout (16x16 f32):    VGPR j: M=j (lanes 0-15) / M=j+8 (lanes 16-31),
//                            N = lane%16.
// B element (K=k, N=n) of X@W^T is W[n][k]  → contiguous 8B read of W row n.
// ---------------------------------------------------------------------------
__global__ void __launch_bounds__(256)
kgat_bigemm(const float* __restrict__ h,
            const float* __restrict__ hn,
            const float* __restrict__ W1,
            const float* __restrict__ W2,
            float* __restrict__ out) {
    __shared__ float sW[2][D * LDS_STRIDE];   // 132 KB

    const int tid = threadIdx.x;

    // --- cooperative staging of W1/W2 into LDS (coalesced b128 reads) ------
    // 128 rows x 32 float4 per matrix = 4096 float4; 256 threads x 16 iters.
    #pragma unroll 4
    for (int i = 0; i < 16; ++i) {
        const int idx  = tid + i * 256;       // 0..4095
        const int row  = idx >> 5;            // 0..127
        const int col4 = idx & 31;            // 0..31
        const float4 a = *(const float4*)(W1 + (size_t)row * D + col4 * 4);
        const float4 b = *(const float4*)(W2 + (size_t)row * D + col4 * 4);
        float* p0 = &sW[0][row * LDS_STRIDE + col4 * 4];
        float* p1 = &sW[1][row * LDS_STRIDE + col4 * 4];
        p0[0] = a.x; p0[1] = a.y; p0[2] = a.z; p0[3] = a.w;
        p1[0] = b.x; p1[1] = b.y; p1[2] = b.z; p1[3] = b.w;
    }
    __syncthreads();

    const int lane  = tid & 31;
    const int wib   = tid >> 5;                              // wave in block
    const int tile  = blockIdx.x * (blockDim.x >> 5) + wib;  // wave-uniform

    if (tile < N_TILES) {                                    // uniform guard
        const int rowbase = tile * 16;
        const int mrow    = rowbase + (lane & 15);  // A-fragment row
        const int khalf   = (lane >> 4) * 2;        // 0 or 2 per lane half
        const int ncol    = lane & 15;              // B/D column in N-tile

        const float* hrow  = h  + (size_t)mrow * D + khalf;
        const float* hnrow = hn + (size_t)mrow * D + khalf;

        for (int nt = 0; nt < 8; ++nt) {
            const int nbase = nt * 16;
            v8f acc1 = {};
            v8f acc2 = {};

            const int bofs = (nbase + ncol) * LDS_STRIDE + khalf;

            #pragma unroll 8
            for (int ks = 0; ks < 32; ++ks) {
                const int kb = ks * 4;
                v2f hv  = *(const v2f*)(hrow  + kb);
                v2f hnv = *(const v2f*)(hnrow + kb);
                v2f a_add = hv + hnv;
                v2f a_mul = hv * hnv;
                v2f b1 = *(const v2f*)(&sW[0][bofs + kb]);   // ds_load_b64
                v2f b2 = *(const v2f*)(&sW[1][bofs + kb]);   // ds_load_b64

                acc1 = __builtin_amdgcn_wmma_f32_16x16x4_f32(
                    false, a_add, false, b1, (short)0, acc1, false, false);
                acc2 = __builtin_amdgcn_wmma_f32_16x16x4_f32(
                    false, a_mul, false, b2, (short)0, acc2, false, false);
            }

            // leaky_relu on both branches, sum, store 16x16 D tile.
            #pragma unroll
            for (int j = 0; j < 8; ++j) {
                float a = acc1[j];
                float b = acc2[j];
                a = a > 0.f ? a : 0.01f * a;
                b = b > 0.f ? b : 0.01f * b;
                const int row = rowbase + j + ((lane >> 4) << 3);
                out[(size_t)row * D + nbase + ncol] = a + b;
            }
        }
    }
}

// ---------------------------------------------------------------------------
extern "C" void kernel_launch(void* const* d_in, const int* in_sizes, int n_in,
                              void* d_out, int out_size, void* d_ws, size_t ws_size,
                              hipStream_t stream) {
    const float* nfeat = (const float*)d_in[0];   // [50000,128]
    const float* w     = (const float*)d_in[1];   // [640000]
    const float* W1    = (const float*)d_in[2];   // [128,128]
    const float* W2    = (const float*)d_in[3];   // [128,128]
    const int*   src   = (const int*)d_in[4];     // [640000]
    const int*   dst   = (const int*)d_in[5];     // [640000]

    float* hn  = (float*)d_out;                       // output 0: h_neighbor
    float* out = (float*)d_out + (size_t)N_NODES * D; // output 1: out

    // 1) zero h_neighbor accumulator (6.4M floats = 1.6M float4)
    {
        const int n4 = N_NODES * D / 4;
        kgat_zero<<<(n4 + 255) / 256, 256, 0, stream>>>(hn, n4);
    }

    // 2) edge scatter: one wave per edge, 8 waves per 256-thread block
    {
        const int blocks = (N_EDGES * 32 + 255) / 256;  // 80000
        kgat_scatter<<<blocks, 256, 0, stream>>>(nfeat, w, src, dst, hn);
    }

    // 3) bi-interaction GEMM: one wave per 16-row tile, 8 waves per block
    {
        const int blocks = (N_TILES + 7) / 8;  // 391
        kgat_bigemm<<<blocks, 256, 0, stream>>>(nfeat, hn, W1, W2, out);
    }
}